// ClusterEntropyModel_49289044689483
// MI455X (gfx1250) — compile-verified
//
#include <hip/hip_runtime.h>
#include <hip/hip_bf16.h>

// ---------------------------------------------------------------------------
// ClusterEntropyModel for MI455X (gfx1250, wave32).
// Phase 1: per-cluster min/max via LDS atomics on sortable-uint floats.
// Phase 2: per-cluster truncated softmax (16 blocks x 256 threads).
// Phase 3: streaming bits pass: 16KB prob table staged to LDS via
//          GLOBAL_LOAD_ASYNC_TO_LDS (ASYNCcnt), packed float4 param gathers,
//          WMMA (v_wmma_f32_16x16x4_f32) wave reduction.
// Phase 4: deterministic final reduce of per-block partials.
// ---------------------------------------------------------------------------

#define NUM_C 16
#define NUM_L 256
#define MAXBLOCKS 2048

typedef __attribute__((ext_vector_type(2))) float v2f;
typedef __attribute__((ext_vector_type(8))) float v8f;
typedef __attribute__((ext_vector_type(4))) int   v4i;

#if __has_builtin(__builtin_amdgcn_global_load_async_to_lds_b128) && \
    __has_builtin(__builtin_amdgcn_s_wait_asynccnt)
#define HAVE_ASYNC_LDS 1
typedef __attribute__((address_space(1))) v4i gas_v4i;   // global (AS1) int4
typedef __attribute__((address_space(3))) v4i las_v4i;   // LDS (AS3) int4
#endif

__device__ __forceinline__ unsigned encf(float f) {
    unsigned u = __float_as_uint(f);
    return u ^ ((unsigned)((int)u >> 31) | 0x80000000u);
}
__device__ __forceinline__ float decf(unsigned e) {
    unsigned mask = (e & 0x80000000u) ? 0x80000000u : 0xFFFFFFFFu;
    return __uint_as_float(e ^ mask);
}

// ---------------------------------------------------------------------------
__global__ void cem_init_kernel(unsigned* wminE, unsigned* wmaxE) {
    int t = threadIdx.x;
    if (t < NUM_C) {
        wminE[t] = 0xFFFFFFFFu;   // encoded max sortable value
        wmaxE[t] = 0x00000000u;   // encoded min sortable value
    }
}

// ---------------------------------------------------------------------------
__global__ void cem_minmax_kernel(const float* __restrict__ qw,
                                  const int* __restrict__ ca,
                                  int N, unsigned* gmin, unsigned* gmax) {
    __shared__ unsigned smin[NUM_C], smax[NUM_C];
    if (threadIdx.x < NUM_C) { smin[threadIdx.x] = 0xFFFFFFFFu; smax[threadIdx.x] = 0u; }
    __syncthreads();

    const int tid = blockIdx.x * blockDim.x + threadIdx.x;
    const int nt  = gridDim.x * blockDim.x;
    const int N4  = N >> 2;
    const float4* qw4 = (const float4*)qw;
    const int4*   ca4 = (const int4*)ca;

    for (int i = tid; i < N4; i += nt) {
        __builtin_prefetch(qw4 + i + nt, 0, 0);
        __builtin_prefetch(ca4 + i + nt, 0, 0);
        float4 w = qw4[i];
        int4   c = ca4[i];
        unsigned e;
        e = encf(w.x); atomicMin(&smin[c.x & (NUM_C-1)], e); atomicMax(&smax[c.x & (NUM_C-1)], e);
        e = encf(w.y); atomicMin(&smin[c.y & (NUM_C-1)], e); atomicMax(&smax[c.y & (NUM_C-1)], e);
        e = encf(w.z); atomicMin(&smin[c.z & (NUM_C-1)], e); atomicMax(&smax[c.z & (NUM_C-1)], e);
        e = encf(w.w); atomicMin(&smin[c.w & (NUM_C-1)], e); atomicMax(&smax[c.w & (NUM_C-1)], e);
    }
    for (int i = (N4 << 2) + tid; i < N; i += nt) {
        unsigned e = encf(qw[i]);
        int c = ca[i] & (NUM_C-1);
        atomicMin(&smin[c], e); atomicMax(&smax[c], e);
    }
    __syncthreads();
    if (threadIdx.x < NUM_C) {
        atomicMin(&gmin[threadIdx.x], smin[threadIdx.x]);
        atomicMax(&gmax[threadIdx.x], smax[threadIdx.x]);
    }
}

// ---------------------------------------------------------------------------
// One block per cluster; blockDim.x == NUM_L == 256.
// Truncated+renormalized softmax collapses to: p[t] = exp(l[t]-m)/sum_{t<nl}.
// Emits packed per-cluster params: (wmin, scale, nlm1, validMask) as float4.
__global__ void cem_probs_kernel(const float* __restrict__ logits,
                                 const int* __restrict__ bitalloc,
                                 const unsigned* __restrict__ wminE,
                                 const unsigned* __restrict__ wmaxE,
                                 float* __restrict__ P,
                                 float4* __restrict__ cparam) {
    __shared__ float red[NUM_L];
    const int c = blockIdx.x;
    const int t = threadIdx.x;

    float l = logits[c * NUM_L + t];
    red[t] = l;
    __syncthreads();
    for (int s = NUM_L / 2; s > 0; s >>= 1) {
        if (t < s) red[t] = fmaxf(red[t], red[t + s]);
        __syncthreads();
    }
    float m = red[0];
    __syncthreads();

    const int nl = 1 << bitalloc[c];
    float e = __expf(l - m);
    red[t] = (t < nl) ? e : 0.0f;
    __syncthreads();
    for (int s = NUM_L / 2; s > 0; s >>= 1) {
        if (t < s) red[t] += red[t + s];
        __syncthreads();
    }
    float S = red[0];
    P[c * NUM_L + t] = (t < nl) ? (e / S) : 0.0f;

    if (t == 0) {
        float wmin = decf(wminE[c]);
        float wmax = decf(wmaxE[c]);
        float rng  = wmax - wmin;
        int   v    = (rng >= 1e-8f) ? 1 : 0;
        float safe = v ? rng : 1.0f;
        float4 prm;
        prm.x = wmin;
        prm.y = (float)(nl - 1) / safe;   // scale
        prm.z = (float)(nl - 1);          // clamp hi
        prm.w = v ? 1.0f : 0.0f;          // valid mask
        cparam[c] = prm;
    }
}

// ---------------------------------------------------------------------------
// Per-element: one ds_load_b128 (params) + adjacent pf/pc loads (read2).
// The idx_c clamp is dropped: alpha>0 implies idx_f+1 <= nl-1, and when
// alpha==0 the pc term contributes exactly 0 (pad keeps row[idx_f+1] finite).
__device__ __forceinline__ float elem_bits(float w, int c,
                                           const float* sP, const float4* sParam) {
    c &= (NUM_C - 1);
    float4 prm = sParam[c];                  // (min, scale, nlm1, valid)
    float soft = (w - prm.x) * prm.y;
    soft = fminf(fmaxf(soft, 0.0f), prm.z);
    int idxf = (int)soft;                    // soft >= 0 so trunc == floor
    float alpha = soft - (float)idxf;
    const float* row = sP + (c << 8);
    float pf = row[idxf];
    float pc = row[idxf + 1];
    float interp = fmaf(alpha, pc - pf, pf); // (1-a)*pf + a*pc
    return -__log2f(interp + 1e-10f) * prm.w;
}

__global__ void __launch_bounds__(256)
cem_main_kernel(const float* __restrict__ qw,
                const int* __restrict__ ca,
                const float* __restrict__ P,
                const float4* __restrict__ cparam,
                int N, float* __restrict__ partials) {
    __shared__ float  sP[NUM_C * NUM_L + 4]; // 16 KB prob table + zero pad
    __shared__ float4 sParam[NUM_C];         // packed per-cluster params
    __shared__ float  sWave[8];              // 256 threads / wave32 = 8 waves

    // ---- Stage prob table into LDS (CDNA5 async copy when available). ----
#if defined(HAVE_ASYNC_LDS)
    {
        const float4* Ps = (const float4*)P;
        float4*       Pd = (float4*)sP;
        for (int i = threadIdx.x; i < (NUM_C * NUM_L) / 4; i += blockDim.x) {
            __builtin_amdgcn_global_load_async_to_lds_b128(
                (gas_v4i*)(Ps + i), (las_v4i*)(Pd + i),
                /*offset=*/0, /*cpol=*/0);
        }
        __builtin_amdgcn_s_wait_asynccnt(0);
    }
#else
    {
        const float4* Ps = (const float4*)P;
        float4*       Pd = (float4*)sP;
        for (int i = threadIdx.x; i < (NUM_C * NUM_L) / 4; i += blockDim.x) Pd[i] = Ps[i];
    }
#endif
    if (threadIdx.x < NUM_C) sParam[threadIdx.x] = cparam[threadIdx.x];
    if (threadIdx.x == 0) {                  // zero pad for unclamped pc load
        sP[NUM_C * NUM_L + 0] = 0.0f;
        sP[NUM_C * NUM_L + 1] = 0.0f;
        sP[NUM_C * NUM_L + 2] = 0.0f;
        sP[NUM_C * NUM_L + 3] = 0.0f;
    }
    __syncthreads();

    const int tid = blockIdx.x * blockDim.x + threadIdx.x;
    const int nt  = gridDim.x * blockDim.x;
    const int N4  = N >> 2;
    const float4* qw4 = (const float4*)qw;
    const int4*   ca4 = (const int4*)ca;

    float acc0 = 0.0f, acc1 = 0.0f;          // two lane-private accumulators
    for (int i = tid; i < N4; i += nt) {
        __builtin_prefetch(qw4 + i + nt, 0, 0);
        __builtin_prefetch(ca4 + i + nt, 0, 0);
        float4 w = qw4[i];
        int4   c = ca4[i];
        acc0 += elem_bits(w.x, c.x, sP, sParam);
        acc1 += elem_bits(w.y, c.y, sP, sParam);
        acc0 += elem_bits(w.z, c.z, sP, sParam);
        acc1 += elem_bits(w.w, c.w, sP, sParam);
    }
    for (int i = (N4 << 2) + tid; i < N; i += nt) {
        acc0 += elem_bits(qw[i], ca[i], sP, sParam);
    }

    // ---- Wave reduction via V_WMMA_F32_16X16X4_F32 with A = ones. ----
    // B (4x16 f32) holds the wave's 64 partial sums (2 per lane). D = 1*B + 0
    // puts full column sums in every row of D; VGPR0 of D holds column sums
    // in all 32 lanes (rows 0 and 8), so a 32-lane butterfly on D[0] yields
    // 2x the wave total.
    float wave2x;
#if __has_builtin(__builtin_amdgcn_wmma_f32_16x16x4_f32)
    {
        v2f a; a.x = 1.0f; a.y = 1.0f;       // all-ones 16x4 A operand
        v2f b; b.x = acc0; b.y = acc1;
        v8f cc = {};
        cc = __builtin_amdgcn_wmma_f32_16x16x4_f32(
            /*neg_a=*/false, a, /*neg_b=*/false, b,
            /*c_mod=*/(short)0, cc, /*reuse_a=*/false, /*reuse_b=*/false);
        wave2x = cc[0];
        for (int off = 16; off > 0; off >>= 1)
            wave2x += __shfl_xor(wave2x, off, 32);
    }
#else
    {
        float s = acc0 + acc1;
        for (int off = 16; off > 0; off >>= 1)
            s += __shfl_xor(s, off, 32);
        wave2x = s * 2.0f;
    }
#endif
    float wave_sum = wave2x * 0.5f;

    const int wave = threadIdx.x >> 5;
    const int lane = threadIdx.x & 31;
    if (lane == 0) sWave[wave] = wave_sum;
    __syncthreads();
    if (threadIdx.x == 0) {
        float bs = 0.0f;
        #pragma unroll
        for (int i = 0; i < 8; ++i) bs += sWave[i];
        partials[blockIdx.x] = bs;
    }
}

// ---------------------------------------------------------------------------
__global__ void cem_final_kernel(const float* __restrict__ partials, int nb,
                                 float invN, float* __restrict__ out) {
    __shared__ float red[256];
    float s = 0.0f;
    for (int i = threadIdx.x; i < nb; i += 256) s += partials[i];
    red[threadIdx.x] = s;
    __syncthreads();
    for (int k = 128; k > 0; k >>= 1) {
        if (threadIdx.x < k) red[threadIdx.x] += red[threadIdx.x + k];
        __syncthreads();
    }
    if (threadIdx.x == 0) out[0] = red[0] * invN;
}

// ---------------------------------------------------------------------------
extern "C" void kernel_launch(void* const* d_in, const int* in_sizes, int n_in,
                              void* d_out, int out_size, void* d_ws, size_t ws_size,
                              hipStream_t stream) {
    const float* qw     = (const float*)d_in[0];   // quantized_weights [N] f32
    const int*   ca     = (const int*)d_in[1];     // cluster_assignments [N] i32
    const int*   ba     = (const int*)d_in[2];     // bit_allocation [C] i32
    const float* logits = (const float*)d_in[3];   // cluster_logits [C,L] f32
    const int N = in_sizes[0];
    float* out = (float*)d_out;

    // Workspace carve-up (d_ws is 256B-aligned from hipMalloc).
    unsigned char* ws = (unsigned char*)d_ws;
    unsigned* wminE = (unsigned*)ws;  ws += NUM_C * sizeof(unsigned);
    unsigned* wmaxE = (unsigned*)ws;  ws += NUM_C * sizeof(unsigned);
    float*  P      = (float*)ws;      ws += NUM_C * NUM_L * sizeof(float); // 16B aligned
    float4* cparam = (float4*)ws;     ws += NUM_C * sizeof(float4);        // 16B aligned
    float*  partials = (float*)ws;    // MAXBLOCKS floats

    int blocks = (N + 1023) / 1024;   // 256 threads x 4 elems
    if (blocks > MAXBLOCKS) blocks = MAXBLOCKS;
    if (blocks < 1) blocks = 1;

    cem_init_kernel<<<1, 64, 0, stream>>>(wminE, wmaxE);
    cem_minmax_kernel<<<blocks, 256, 0, stream>>>(qw, ca, N, wminE, wmaxE);
    cem_probs_kernel<<<NUM_C, NUM_L, 0, stream>>>(logits, ba, wminE, wmaxE, P, cparam);
    cem_main_kernel<<<blocks, 256, 0, stream>>>(qw, ca, P, cparam, N, partials);
    cem_final_kernel<<<1, 256, 0, stream>>>(partials, blocks, 1.0f / (float)N, out);
}